// NR_GraphAttention_67156108640947
// MI455X (gfx1250) — compile-verified
//
#include <hip/hip_runtime.h>
#include <math.h>

#define N_NODES 100000
#define R_REL   1000
#define T_EDGE  600000
#define DIM     128
#define OSTRIDE 640   // D * (2*DEPTH+1)

typedef __attribute__((ext_vector_type(2))) float v2f;
typedef __attribute__((ext_vector_type(8))) float v8f;

// ---- float atomic max via sign-aware integer atomics (init must be -inf) ----
__device__ __forceinline__ void atomicMaxF(float* a, float v) {
  if (v >= 0.0f) atomicMax((int*)a, __float_as_int(v));
  else           atomicMin((unsigned int*)a, __float_as_uint(v));
}

// ---- output block 0: tanh(features) ----
__global__ void k_tanh_in(const float* __restrict__ f, float* __restrict__ out) {
  int idx = blockIdx.x * blockDim.x + threadIdx.x;
  if (idx >= N_NODES * DIM) return;
  out[(size_t)(idx >> 7) * OSTRIDE + (idx & 127)] = tanhf(f[idx]);
}

// ---- per-relation L2 norm of rel_emb rows ----
__global__ void k_relnorm(const float* __restrict__ rel_emb, float* __restrict__ relnrm) {
  int r = blockIdx.x * blockDim.x + threadIdx.x;
  if (r >= R_REL) return;
  const float4* p = (const float4*)(rel_emb + (size_t)r * DIM);
  float acc = 0.0f;
  #pragma unroll 4
  for (int i = 0; i < DIM / 4; ++i) {
    float4 v = p[i];
    acc += v.x * v.x + v.y * v.y + v.z * v.z + v.w * v.w;
  }
  relnrm[r] = sqrtf(acc);
}

// ---- dotk[l][r] = rel_emb[r] . attn_kernels[l]  via V_WMMA_F32_16X16X4_F32 ----
// One wave per 16-relation tile. A = 16x4 f32 (lanes0-15: K0,K1 ; lanes16-31: K2,K3),
// B = 4x16 f32 mirrored layout, columns 0/1 carry the two attention kernels.
// B operand is built branch-free: clamped safe address + per-lane select, so the
// unrolled K-loop contains no EXEC-mask manipulation around the loads.
__global__ void k_dotk_wmma(const float* __restrict__ rel_emb,
                            const float* __restrict__ ak,   // [2*128]
                            float* __restrict__ dotk) {     // [2*R]
  const int NT = (R_REL + 15) / 16;                        // 63 tiles
  int tile = blockIdx.x * (blockDim.x >> 5) + (threadIdx.x >> 5);
  if (tile >= NT) return;                                  // wave-uniform exit
  int lane  = threadIdx.x & 31;
  int mrow  = lane & 15;
  int khalf = (lane < 16) ? 0 : 2;
  int rr    = tile * 16 + mrow;
  int rsafe = rr < R_REL ? rr : (R_REL - 1);
  const float* arow = rel_emb + (size_t)rsafe * DIM;
  // safe B row: lanes with N-col >= 2 read column 0's data but select 0.0f
  const float* brow = ak + (size_t)((mrow < 2) ? mrow : 0) * DIM;
  const float  bmsk = (mrow < 2) ? 1.0f : 0.0f;

  v8f c = {};
  #pragma unroll
  for (int k0 = 0; k0 < DIM; k0 += 4) {
    v2f a = *(const v2f*)(arow + k0 + khalf);   // 8B aligned
    v2f b = *(const v2f*)(brow + k0 + khalf);
    b.x *= bmsk;                                // v_cndmask/mul, no branch
    b.y *= bmsk;
    c = __builtin_amdgcn_wmma_f32_16x16x4_f32(false, a, false, b,
                                              (short)0, c, false, false);
  }
  // D layout: lanes0-15 -> N=lane, M=v ; lanes16-31 -> N=lane-16, M=v+8
  if (mrow <= 1) {
    int base = tile * 16 + ((lane < 16) ? 0 : 8);
    float* dst = dotk + (size_t)mrow * R_REL;
    #pragma unroll
    for (int v = 0; v < 8; ++v) {
      int ri = base + v;
      if (ri < R_REL) dst[ri] = c[v];
    }
  }
}

// ---- per-layer init: zero output block, m = -inf, s = 0 ----
__global__ void k_init_layer(float* __restrict__ outb, float* __restrict__ m,
                             float* __restrict__ s) {
  int idx = blockIdx.x * blockDim.x + threadIdx.x;
  if (idx < N_NODES * DIM)
    outb[(size_t)(idx >> 7) * OSTRIDE + (idx & 127)] = 0.0f;
  if (idx < N_NODES) { m[idx] = -INFINITY; s[idx] = 0.0f; }
}

// ---- relation-layer logits + segment max ----
__global__ void k_edge_logit_rel(const int* __restrict__ reli, const float* __restrict__ rval,
                                 const float* __restrict__ relnrm, const float* __restrict__ dotk_l,
                                 const int* __restrict__ row,
                                 float* __restrict__ logit, float* __restrict__ m) {
  int t = blockIdx.x * blockDim.x + threadIdx.x;
  if (t >= T_EDGE) return;
  int   re = reli[t];
  float rv = rval[t];
  float sc = rv / fmaxf(rv * relnrm[re], 1e-12f);
  float L  = sc * dotk_l[re];
  logit[t] = L;
  atomicMaxF(&m[row[t]], L);
}

// ---- exp-sum for segment softmax ----
__global__ void k_exp_sum(const float* __restrict__ logit, const int* __restrict__ row,
                          const float* __restrict__ m, float* __restrict__ s) {
  int t = blockIdx.x * blockDim.x + threadIdx.x;
  if (t >= T_EDGE) return;
  int r = row[t];
  atomicAdd(&s[r], expf(logit[t] - m[r]));
}

// ---- relation-layer aggregation: Householder reflect + weighted scatter-add ----
// One wave32 per edge, 4 floats per lane (float4).
__global__ void k_aggr_rel(const float* __restrict__ fin, const float* __restrict__ rel_emb,
                           const int* __restrict__ reli, const float* __restrict__ rval,
                           const float* __restrict__ relnrm,
                           const int* __restrict__ row, const int* __restrict__ col,
                           const float* __restrict__ logit, const float* __restrict__ m,
                           const float* __restrict__ s, float* __restrict__ outb) {
  int t = blockIdx.x * (blockDim.x >> 5) + (threadIdx.x >> 5);
  if (t >= T_EDGE) return;
  int lane = threadIdx.x & 31;
  int r = row[t], c = col[t], re = reli[t];
  float rv = rval[t];
  float sc = rv / fmaxf(rv * relnrm[re], 1e-12f);

  float4 nf = *(const float4*)(fin + (size_t)c * OSTRIDE + lane * 4);
  float4 u  = *(const float4*)(rel_emb + (size_t)re * DIM + lane * 4);
  u.x *= sc; u.y *= sc; u.z *= sc; u.w *= sc;

  float dot = nf.x * u.x + nf.y * u.y + nf.z * u.z + nf.w * u.w;
  #pragma unroll
  for (int off = 16; off; off >>= 1) dot += __shfl_xor(dot, off, 32);

  float coeff = expf(logit[t] - m[r]) / s[r];
  float tw = -2.0f * dot;
  float* dst = outb + (size_t)r * OSTRIDE + lane * 4;
  atomicAdd(dst + 0, (nf.x + tw * u.x) * coeff);
  atomicAdd(dst + 1, (nf.y + tw * u.y) * coeff);
  atomicAdd(dst + 2, (nf.z + tw * u.z) * coeff);
  atomicAdd(dst + 3, (nf.w + tw * u.w) * coeff);
}

// ---- high-layer logits: per-edge 128-dim dot with high_atts[l] ----
__global__ void k_edge_logit_high(const float* __restrict__ fin, const float* __restrict__ ha,
                                  const int* __restrict__ hrow, const int* __restrict__ hcol,
                                  float* __restrict__ logit, float* __restrict__ m) {
  int t = blockIdx.x * (blockDim.x >> 5) + (threadIdx.x >> 5);
  if (t >= T_EDGE) return;
  int lane = threadIdx.x & 31;
  int c = hcol[t];
  float4 nf = *(const float4*)(fin + (size_t)c * OSTRIDE + lane * 4);
  float4 a4 = *(const float4*)(ha + lane * 4);
  float dot = nf.x * a4.x + nf.y * a4.y + nf.z * a4.z + nf.w * a4.w;
  #pragma unroll
  for (int off = 16; off; off >>= 1) dot += __shfl_xor(dot, off, 32);
  if (lane == 0) {
    logit[t] = dot;
    atomicMaxF(&m[hrow[t]], dot);
  }
}

// ---- high-layer aggregation ----
__global__ void k_aggr_high(const float* __restrict__ fin,
                            const int* __restrict__ hrow, const int* __restrict__ hcol,
                            const float* __restrict__ logit, const float* __restrict__ m,
                            const float* __restrict__ s, float* __restrict__ outb) {
  int t = blockIdx.x * (blockDim.x >> 5) + (threadIdx.x >> 5);
  if (t >= T_EDGE) return;
  int lane = threadIdx.x & 31;
  int r = hrow[t], c = hcol[t];
  float coeff = expf(logit[t] - m[r]) / s[r];
  float4 nf = *(const float4*)(fin + (size_t)c * OSTRIDE + lane * 4);
  float* dst = outb + (size_t)r * OSTRIDE + lane * 4;
  atomicAdd(dst + 0, nf.x * coeff);
  atomicAdd(dst + 1, nf.y * coeff);
  atomicAdd(dst + 2, nf.z * coeff);
  atomicAdd(dst + 3, nf.w * coeff);
}

// ---- in-place tanh over an output block ----
__global__ void k_tanh_block(float* __restrict__ outb) {
  int idx = blockIdx.x * blockDim.x + threadIdx.x;
  if (idx >= N_NODES * DIM) return;
  size_t o = (size_t)(idx >> 7) * OSTRIDE + (idx & 127);
  outb[o] = tanhf(outb[o]);
}

extern "C" void kernel_launch(void* const* d_in, const int* in_sizes, int n_in,
                              void* d_out, int out_size, void* d_ws, size_t ws_size,
                              hipStream_t stream) {
  const float* features = (const float*)d_in[0];
  const float* rel_emb  = (const float*)d_in[1];
  const float* r_val    = (const float*)d_in[2];
  const float* attn_k   = (const float*)d_in[3];   // [2,128]
  const float* high_att = (const float*)d_in[4];   // [2,128]
  const int*   adj      = (const int*)d_in[5];     // [2,T]
  const int*   r_index  = (const int*)d_in[6];     // [2,T] (row 0 == arange)
  const int*   high_nei = (const int*)d_in[7];     // [2,T]
  float* out = (float*)d_out;

  float* ws     = (float*)d_ws;
  float* logit  = ws;                              // T
  float* m      = ws + T_EDGE;                     // N
  float* s      = ws + T_EDGE + N_NODES;           // N
  float* relnrm = ws + T_EDGE + 2 * N_NODES;       // R
  float* dotk   = relnrm + R_REL;                  // 2*R

  const int* row  = adj;
  const int* col  = adj + T_EDGE;
  const int* reli = r_index + T_EDGE;
  const int* hrow = high_nei;
  const int* hcol = high_nei + T_EDGE;

  const int ND_BLOCKS   = (N_NODES * DIM + 255) / 256;       // 50000
  const int EDGE_BLOCKS = (T_EDGE + 255) / 256;              // 2344
  const int WAVE_BLOCKS = (T_EDGE + 7) / 8;                  // 75000 (8 waves/block)

  // block 0 + per-relation precompute (WMMA)
  k_tanh_in<<<ND_BLOCKS, 256, 0, stream>>>(features, out);
  k_relnorm<<<(R_REL + 255) / 256, 256, 0, stream>>>(rel_emb, relnrm);
  k_dotk_wmma<<<8, 256, 0, stream>>>(rel_emb, attn_k, dotk);

  int off_in = 0;
  // ---- two relation-attention layers ----
  for (int l = 0; l < 2; ++l) {
    int off_out = (l + 1) * DIM;
    k_init_layer<<<ND_BLOCKS, 256, 0, stream>>>(out + off_out, m, s);
    k_edge_logit_rel<<<EDGE_BLOCKS, 256, 0, stream>>>(reli, r_val, relnrm,
                                                      dotk + (size_t)l * R_REL,
                                                      row, logit, m);
    k_exp_sum<<<EDGE_BLOCKS, 256, 0, stream>>>(logit, row, m, s);
    k_aggr_rel<<<WAVE_BLOCKS, 256, 0, stream>>>(out + off_in, rel_emb, reli, r_val,
                                                relnrm, row, col, logit, m, s,
                                                out + off_out);
    k_tanh_block<<<ND_BLOCKS, 256, 0, stream>>>(out + off_out);
    off_in = off_out;
  }
  // ---- two high-neighbor attention layers ----
  for (int l = 0; l < 2; ++l) {
    int off_out = (3 + l) * DIM;
    k_init_layer<<<ND_BLOCKS, 256, 0, stream>>>(out + off_out, m, s);
    k_edge_logit_high<<<WAVE_BLOCKS, 256, 0, stream>>>(out + off_in,
                                                       high_att + (size_t)l * DIM,
                                                       hrow, hcol, logit, m);
    k_exp_sum<<<EDGE_BLOCKS, 256, 0, stream>>>(logit, hrow, m, s);
    k_aggr_high<<<WAVE_BLOCKS, 256, 0, stream>>>(out + off_in, hrow, hcol,
                                                 logit, m, s, out + off_out);
    k_tanh_block<<<ND_BLOCKS, 256, 0, stream>>>(out + off_out);
    off_in = off_out;
  }
}